// CrossLayer_51831665328222
// MI455X (gfx1250) — compile-verified
//
#include <hip/hip_runtime.h>

// DCN cross layer, algebraically collapsed:
//   out = x * t4 + sumB,   t_k = t_{k-1} + t_{k-1}*a_k + d_k,
//   a_k = x . w_k (per row, one fused pass), d_k = (b_1+..+b_{k-1}) . w_k (scalars).
// Memory-bound: 64MB read + 64MB write @ 23.3 TB/s ~ 5.5us.

#define DDIM 1024

typedef __attribute__((ext_vector_type(2))) float v2f;
typedef __attribute__((ext_vector_type(4))) float v4f;
typedef __attribute__((ext_vector_type(8))) float v8f;
typedef __attribute__((ext_vector_type(4))) int   v4i;

typedef __attribute__((address_space(1))) v4i glob_v4i;  // global (device) AS
typedef __attribute__((address_space(3))) v4i lds_v4i;   // LDS (shared) AS

#if defined(__gfx1250__) && __has_builtin(__builtin_amdgcn_global_load_async_to_lds_b128)
#define HAVE_ASYNC_LDS 1
#else
#define HAVE_ASYNC_LDS 0
#endif

#if defined(__gfx1250__) && __has_builtin(__builtin_amdgcn_wmma_f32_16x16x4_f32)
#define HAVE_WMMA_F32 1
#else
#define HAVE_WMMA_F32 0
#endif

__device__ __forceinline__ float wave_reduce_add(float v) {
#pragma unroll
  for (int m = 16; m >= 1; m >>= 1) v += __shfl_xor(v, m, 32);
  return v;
}

// ---------------------------------------------------------------------------
// Precompute kernel (1 wave): ws[0..3] = d_1..d_4, ws[4..1027] = sumB[0..1023]
// d_k computed via V_WMMA_F32_16X16X4_F32: C[i][k] = b_i . w_k (i,k < 4),
// accumulated over K=1024 in 256 steps; d_{k+1} = sum_{i<k} C[i][k] in lane k.
// ---------------------------------------------------------------------------
__global__ __launch_bounds__(32) void dcn_precompute_kernel(
    const float* __restrict__ w, const float* __restrict__ b,
    float* __restrict__ ws) {
  const int lane = threadIdx.x & 31;

  // sumB = b1 + b2 + b3 + b4 (1024 floats)
#pragma unroll
  for (int j = 0; j < 8; ++j) {
    const int e = j * 128 + lane * 4;
    v4f b0 = *(const v4f*)(b + 0 * DDIM + e);
    v4f b1 = *(const v4f*)(b + 1 * DDIM + e);
    v4f b2 = *(const v4f*)(b + 2 * DDIM + e);
    v4f b3 = *(const v4f*)(b + 3 * DDIM + e);
    v4f s = b0 + b1 + b2 + b3;
    *(v4f*)(ws + 4 + e) = s;
  }

#if HAVE_WMMA_F32
  // 16x4 A layout: lanes 0-15 hold K=0(V0),K=1(V1); lanes 16-31 hold K=2,K=3.
  // 4x16 B layout mirrors: V0 = {K0 | K2}, V1 = {K1 | K3}, N = lane%16.
  const int m = lane & 15;          // A row / B col
  const int h = lane >> 4;          // K-half selector
  const float msk = (m < 4) ? 1.0f : 0.0f;
  const int rb = (m < 4 ? m : 0) * DDIM;
  v8f acc = {};
  for (int c = 0; c < 256; ++c) {
    const int col = c * 4 + h * 2;
    v2f A, Bv;
    A.x  = msk * b[rb + col];
    A.y  = msk * b[rb + col + 1];
    Bv.x = msk * w[rb + col];
    Bv.y = msk * w[rb + col + 1];
    acc = __builtin_amdgcn_wmma_f32_16x16x4_f32(
        false, A, false, Bv, (short)0, acc, false, false);
  }
  // C/D layout: VGPR i, lanes 0-15 -> M=i, N=lane. d_{n+1} = sum_{i<n} C[i][n].
  if (lane < 4) {
    float dd = 0.0f;
    if (lane > 0) dd += acc[0];
    if (lane > 1) dd += acc[1];
    if (lane > 2) dd += acc[2];
    ws[lane] = dd;
  }
#else
  float q2 = 0.f, q3 = 0.f, q4 = 0.f;
  for (int e = lane; e < DDIM; e += 32) {
    const float b1v = b[e], b2v = b[DDIM + e], b3v = b[2 * DDIM + e];
    const float pb1 = b1v, pb2 = pb1 + b2v, pb3 = pb2 + b3v;
    q2 += pb1 * w[1 * DDIM + e];
    q3 += pb2 * w[2 * DDIM + e];
    q4 += pb3 * w[3 * DDIM + e];
  }
  q2 = wave_reduce_add(q2);
  q3 = wave_reduce_add(q3);
  q4 = wave_reduce_add(q4);
  if (lane == 0) { ws[0] = 0.f; ws[1] = q2; ws[2] = q3; ws[3] = q4; }
#endif
}

// ---------------------------------------------------------------------------
// Main kernel: 8 wave32 per block, one row per wave. x row lives in VGPRs
// (8 x b128); w + sumB staged to LDS via async global->LDS (20KB/WG).
// ---------------------------------------------------------------------------
__global__ __launch_bounds__(256) void dcn_cross_kernel(
    const float* __restrict__ x, const float* __restrict__ w,
    const float* __restrict__ ws, float* __restrict__ out, int nrows) {
  __shared__ float lds[5 * DDIM];  // [0,4D): w rows, [4D,5D): sumB
  const int tid = threadIdx.x;

#if HAVE_ASYNC_LDS
#pragma unroll
  for (int i = 0; i < 4; ++i) {
    const int off = (tid + i * 256) * 4;  // 16KB of weights
    __builtin_amdgcn_global_load_async_to_lds_b128(
        (glob_v4i*)(w + off), (lds_v4i*)(lds + off), 0, 0);
  }
  {
    const int off = tid * 4;  // 4KB sumB
    __builtin_amdgcn_global_load_async_to_lds_b128(
        (glob_v4i*)(ws + 4 + off), (lds_v4i*)(lds + 4 * DDIM + off), 0, 0);
  }
#if __has_builtin(__builtin_amdgcn_s_wait_asynccnt)
  __builtin_amdgcn_s_wait_asynccnt(0);
#else
  asm volatile("s_wait_asynccnt 0" ::: "memory");
#endif
#else
#pragma unroll
  for (int i = 0; i < 4; ++i) {
    const int off = (tid + i * 256) * 4;
    *(v4f*)(lds + off) = *(const v4f*)(w + off);
  }
  *(v4f*)(lds + 4 * DDIM + tid * 4) = *(const v4f*)(ws + 4 + tid * 4);
#endif
  const float d1 = ws[0], d2 = ws[1], d3 = ws[2], d4 = ws[3];
  __syncthreads();

  const int wave = tid >> 5, lane = tid & 31;
  const int row = blockIdx.x * 8 + wave;
  if (row >= nrows) return;

  const float* xr = x + (size_t)row * DDIM;
  v4f xv[8];
#pragma unroll
  for (int j = 0; j < 8; ++j)
    xv[j] = __builtin_nontemporal_load((const v4f*)(xr + j * 128 + lane * 4));

  // Four dot partials in one fused pass
  float pa0 = 0.f, pa1 = 0.f, pa2 = 0.f, pa3 = 0.f;
#pragma unroll
  for (int j = 0; j < 8; ++j) {
    const int off = j * 128 + lane * 4;
    v4f w0 = *(v4f*)(lds + 0 * DDIM + off);
    v4f w1 = *(v4f*)(lds + 1 * DDIM + off);
    v4f w2 = *(v4f*)(lds + 2 * DDIM + off);
    v4f w3 = *(v4f*)(lds + 3 * DDIM + off);
    pa0 += xv[j].x * w0.x + xv[j].y * w0.y + xv[j].z * w0.z + xv[j].w * w0.w;
    pa1 += xv[j].x * w1.x + xv[j].y * w1.y + xv[j].z * w1.z + xv[j].w * w1.w;
    pa2 += xv[j].x * w2.x + xv[j].y * w2.y + xv[j].z * w2.z + xv[j].w * w2.w;
    pa3 += xv[j].x * w3.x + xv[j].y * w3.y + xv[j].z * w3.z + xv[j].w * w3.w;
  }
  pa0 = wave_reduce_add(pa0);
  pa1 = wave_reduce_add(pa1);
  pa2 = wave_reduce_add(pa2);
  pa3 = wave_reduce_add(pa3);

  // t_k = t_{k-1} + t_{k-1}*a_k + d_k
  float t = 1.0f;
  t = __builtin_fmaf(t, pa0, t) + d1;
  t = __builtin_fmaf(t, pa1, t) + d2;
  t = __builtin_fmaf(t, pa2, t) + d3;
  t = __builtin_fmaf(t, pa3, t) + d4;

  float* outr = out + (size_t)row * DDIM;
#pragma unroll
  for (int j = 0; j < 8; ++j) {
    const int off = j * 128 + lane * 4;
    v4f sb = *(v4f*)(lds + 4 * DDIM + off);
    v4f o;
    o.x = __builtin_fmaf(xv[j].x, t, sb.x);
    o.y = __builtin_fmaf(xv[j].y, t, sb.y);
    o.z = __builtin_fmaf(xv[j].z, t, sb.z);
    o.w = __builtin_fmaf(xv[j].w, t, sb.w);
    __builtin_nontemporal_store(o, (v4f*)(outr + off));
  }
}

extern "C" void kernel_launch(void* const* d_in, const int* in_sizes, int n_in,
                              void* d_out, int out_size, void* d_ws, size_t ws_size,
                              hipStream_t stream) {
  const float* x = (const float*)d_in[0];
  const float* w = (const float*)d_in[1];   // (4, 1024, 1) contiguous
  const float* b = (const float*)d_in[2];   // (4, 1024, 1) contiguous
  float* out = (float*)d_out;
  float* ws = (float*)d_ws;                 // ws[0..3]=d_k, ws[4..1027]=sumB

  const int nrows = in_sizes[0] / DDIM;     // 16384
  dcn_precompute_kernel<<<1, 32, 0, stream>>>(w, b, ws);
  dcn_cross_kernel<<<(nrows + 7) / 8, 256, 0, stream>>>(x, w, ws, out, nrows);
  (void)n_in; (void)out_size; (void)ws_size;
}